// MaxLocalActivation_506806141061
// MI455X (gfx1250) — compile-verified
//
#include <hip/hip_runtime.h>
#include <stdint.h>

#define B_  8
#define F_  64
#define BF  512           // B*F
#define N_  10000
#define M1_ 17
#define M2_ 65
#define NPB 32            // nodes per block in gather kernel

// ---------------------------------------------------------------------------
// CDNA5 (gfx1250) async global->LDS helpers.
// Generic pointers to __shared__ carry the workgroup-relative LDS byte offset
// in their low 32 bits (flat-aperture encoding), which is exactly what the
// VDST operand of global_load_async_to_lds_* expects.
// ---------------------------------------------------------------------------
__device__ __forceinline__ uint32_t lds_off_of(const void* p) {
  return (uint32_t)(uintptr_t)p;
}
__device__ __forceinline__ void async_g2l_b32(uint32_t lds_off, const void* g) {
  asm volatile("global_load_async_to_lds_b32 %0, %1, off"
               :: "v"(lds_off), "v"((uint64_t)(uintptr_t)g)
               : "memory");
}
__device__ __forceinline__ void wait_async0() {
  asm volatile("s_wait_asynccnt 0" ::: "memory");
}

// ---------------------------------------------------------------------------
// Kernel 1: transpose x [BF][N] -> xt [N][BF] (tiled 32x32 through LDS).
// Loads staged with async global->LDS copies.
// ---------------------------------------------------------------------------
__global__ __launch_bounds__(256) void transpose_x_kernel(
    const float* __restrict__ x, float* __restrict__ xt) {
  __shared__ float tile[32][33];
  const int tx = threadIdx.x;       // 0..31
  const int ty = threadIdx.y;       // 0..7
  const int n0 = blockIdx.x * 32;   // node tile
  const int bf0 = blockIdx.y * 32;  // channel tile
  const int n = n0 + tx;

  if (n < N_) {
#pragma unroll
    for (int r = 0; r < 4; ++r) {
      const int bfrel = ty + r * 8;
      async_g2l_b32(lds_off_of(&tile[bfrel][tx]),
                    x + (size_t)(bf0 + bfrel) * N_ + n);
    }
  }
  wait_async0();
  __syncthreads();

#pragma unroll
  for (int r = 0; r < 4; ++r) {
    const int nrel = ty + r * 8;
    const int n2 = n0 + nrel;
    if (n2 < N_) {
      xt[(size_t)n2 * BF + bf0 + tx] = tile[tx][nrel];
    }
  }
}

// ---------------------------------------------------------------------------
// Kernel 2: per-node gather + max + weighted combine.
// Block = 512 threads (16 waves), owns NPB=32 nodes.
//   - neighbor index lists async-copied to LDS once per block
//   - 4 groups of 128 threads; group g handles nodes g*8..g*8+7
//   - each thread covers 4 channels (float4): each gather is a contiguous
//     2KB row segment per group -> perfectly coalesced
//   - indices are wave-uniform -> readfirstlane => scalar-base addressing
// ---------------------------------------------------------------------------
__global__ __launch_bounds__(512, 1) void gather_max_kernel(
    const float* __restrict__ xt,    // [N][BF]
    const int*   __restrict__ nbr1,  // [N][M1]
    const int*   __restrict__ nbr2,  // [N][M2]
    const float* __restrict__ wt,    // [3]
    float*       __restrict__ out) { // [BF][N]
  __shared__ int sidx1[NPB * M1_];   // 544 * 4B
  __shared__ int sidx2[NPB * M2_];   // 2080 * 4B

  const int t  = threadIdx.x;
  const int n0 = blockIdx.x * NPB;
  const int nvalid = (N_ - n0 < NPB) ? (N_ - n0) : NPB;

  // Stage neighbor indices (int32) into LDS with async copies.
  const int tot1 = nvalid * M1_;
  for (int i = t; i < tot1; i += 512) {
    const int nl = i / M1_, m = i - nl * M1_;
    async_g2l_b32(lds_off_of(&sidx1[i]), nbr1 + (size_t)(n0 + nl) * M1_ + m);
  }
  const int tot2 = nvalid * M2_;
  for (int i = t; i < tot2; i += 512) {
    const int nl = i / M2_, m = i - nl * M2_;
    async_g2l_b32(lds_off_of(&sidx2[i]), nbr2 + (size_t)(n0 + nl) * M2_ + m);
  }
  wait_async0();
  __syncthreads();

  const float w0 = wt[0], w1 = wt[1], w2 = wt[2];

  const int grp  = t >> 7;       // 0..3 (each 4 waves)
  const int l128 = t & 127;      // lane within group; channel = 4*l128
  const float4* __restrict__ xt4 = (const float4*)xt;  // row = 128 float4

#pragma unroll 1
  for (int k = 0; k < 8; ++k) {
    const int nl = grp * 8 + k;
    const int n  = n0 + nl;
    if (n >= N_) break;

    const float4 self = xt4[(size_t)n * 128 + l128];

    float4 a1 = {-__builtin_inff(), -__builtin_inff(),
                 -__builtin_inff(), -__builtin_inff()};
#pragma unroll
    for (int m = 0; m < M1_; ++m) {
      const int idx = __builtin_amdgcn_readfirstlane(sidx1[nl * M1_ + m]);
      const float4 v = xt4[(size_t)idx * 128 + l128];
      a1.x = fmaxf(a1.x, v.x); a1.y = fmaxf(a1.y, v.y);
      a1.z = fmaxf(a1.z, v.z); a1.w = fmaxf(a1.w, v.w);
    }

    float4 a2 = {-__builtin_inff(), -__builtin_inff(),
                 -__builtin_inff(), -__builtin_inff()};
#pragma unroll
    for (int m = 0; m < M2_; ++m) {
      const int idx = __builtin_amdgcn_readfirstlane(sidx2[nl * M2_ + m]);
      const float4 v = xt4[(size_t)idx * 128 + l128];
      a2.x = fmaxf(a2.x, v.x); a2.y = fmaxf(a2.y, v.y);
      a2.z = fmaxf(a2.z, v.z); a2.w = fmaxf(a2.w, v.w);
    }

    float4 o;
    o.x = w0 * self.x + w1 * a1.x + w2 * a2.x;
    o.y = w0 * self.y + w1 * a1.y + w2 * a2.y;
    o.z = w0 * self.z + w1 * a1.z + w2 * a2.z;
    o.w = w0 * self.w + w1 * a1.w + w2 * a2.w;

    // out is [BF][N]: 4B scatter along n, merged to full lines in L2 since
    // each 64B line is covered by threads of this same block.
    const size_t base = (size_t)(4 * l128) * N_ + n;
    out[base]           = o.x;
    out[base + N_]      = o.y;
    out[base + 2 * N_]  = o.z;
    out[base + 3 * N_]  = o.w;
  }
}

// ---------------------------------------------------------------------------
// Fallback (only if workspace is too small for the transposed copy): direct,
// uncoalesced but correct.
// ---------------------------------------------------------------------------
__global__ void gather_max_fallback_kernel(
    const float* __restrict__ x, const int* __restrict__ nbr1,
    const int* __restrict__ nbr2, const float* __restrict__ wt,
    float* __restrict__ out) {
  const size_t i = (size_t)blockIdx.x * 256 + threadIdx.x;
  if (i >= (size_t)BF * N_) return;
  const int n  = (int)(i % N_);
  const int bf = (int)(i / N_);
  const float* xr = x + (size_t)bf * N_;
  float a1 = -__builtin_inff();
  for (int m = 0; m < M1_; ++m) a1 = fmaxf(a1, xr[nbr1[(size_t)n * M1_ + m]]);
  float a2 = -__builtin_inff();
  for (int m = 0; m < M2_; ++m) a2 = fmaxf(a2, xr[nbr2[(size_t)n * M2_ + m]]);
  out[i] = wt[0] * xr[n] + wt[1] * a1 + wt[2] * a2;
}

extern "C" void kernel_launch(void* const* d_in, const int* in_sizes, int n_in,
                              void* d_out, int out_size, void* d_ws, size_t ws_size,
                              hipStream_t stream) {
  (void)in_sizes; (void)n_in; (void)out_size;
  const float* x    = (const float*)d_in[0];
  const float* wt   = (const float*)d_in[1];
  const int*   nbr1 = (const int*)d_in[2];
  const int*   nbr2 = (const int*)d_in[3];
  float*       out  = (float*)d_out;

  const size_t xt_bytes = (size_t)N_ * BF * sizeof(float);
  if (ws_size >= xt_bytes) {
    float* xt = (float*)d_ws;
    dim3 tb(32, 8);
    dim3 tg((N_ + 31) / 32, BF / 32);
    transpose_x_kernel<<<tg, tb, 0, stream>>>(x, xt);

    const int nblocks = (N_ + NPB - 1) / NPB;  // 313
    gather_max_kernel<<<nblocks, 512, 0, stream>>>(xt, nbr1, nbr2, wt, out);
  } else {
    const size_t total = (size_t)BF * N_;
    const int nblocks = (int)((total + 255) / 256);
    gather_max_fallback_kernel<<<nblocks, 256, 0, stream>>>(x, nbr1, nbr2, wt, out);
  }
}